// ConvolutionalSocialPooling_3582002724870
// MI455X (gfx1250) — compile-verified
//
#include <hip/hip_runtime.h>
#include <hip/hip_bf16.h>
#include <float.h>

// ---------------------------------------------------------------------------
// Convolutional Social Pooling, fused for MI455X (gfx1250, wave32).
//
// Kernel 1 (96 threads = 3 waves per batch element):
//   stage 1: winner-per-cell via LDS atomicMax
//   stage 2: gather winner features into a PRE-PADDED, TRANSPOSED LDS grid
//            socPT[(h+dh)*3+w][i]  (45 rows x 128 ch, rows 0..2 / 42..44 = 0)
//            + pre-stage conv31 weights reordered to k' = dh*128+i in LDS
//   stage 3: conv3x1 as GEMM via V_WMMA_F32_16X16X4_F32 with K ordered
//            dh-major: inner loop = 2x ds_load_b64 + 1 wmma, no divides,
//            no padding branches (padding is baked into socPT rows)
//   stage 4: conv1x1 (16->32) ALSO via WMMA (K=16 -> 4 steps, 2 row-tiles)
//   stage 5: 3x3/3 maxpool -> X[b][160] in workspace
//
// Kernel 2: FC GEMM out = lrelu(X @ fc_w^T + fc_b), (8192x160)x(160x64),
//   one 16x16 tile per wave, 40 fp32-WMMA k-steps, unrolled x8.
// ---------------------------------------------------------------------------

typedef float v2f __attribute__((ext_vector_type(2)));
typedef float v8f __attribute__((ext_vector_type(8)));

#define GRID_H 13
#define GRID_W 3
#define NCELL  39          // 13*3
#define NROWP  45          // (13+2 padded H rows) * 3
#define ENC_DIM 128
#define TSTEPS  16
#define C31     16
#define C11     32
#define KNBR    32
#define BATCH   8192
#define FC_DIM  160        // 32 * 5 * 1
#define SOC_D   64

#define SOC_STRIDE 130     // socPT row stride (128 + bank pad)
#define W31_STRIDE 386     // w31A row stride (384 + bank pad)
#define Y16_STRIDE 18      // y16T row stride (16 + bank pad)

__device__ __forceinline__ float lrelu(float x) {
    return x > 0.0f ? x : 0.1f * x;
}

// ---------------------------------------------------------------------------
// Kernel 1: social grid + conv3x1 (WMMA) + conv1x1 (WMMA) + maxpool
// ---------------------------------------------------------------------------
__global__ __launch_bounds__(96)
void csp_grid_conv_kernel(const float* __restrict__ enc,    // (N_VEH,128,16)
                          const int*   __restrict__ nb,     // (B,32)
                          const int*   __restrict__ gp,     // (N_VEH,2)
                          const float* __restrict__ w31,    // (16,128,3,1)
                          const float* __restrict__ b31,    // (16)
                          const float* __restrict__ w11,    // (32,16,1,1)
                          const float* __restrict__ b11,    // (32)
                          float*       __restrict__ X)      // (B,160) workspace
{
    __shared__ int winner[NCELL];
    // B matrix, transposed + H-padded: socPT[row][i], row = (h+dh)*3+w
    __shared__ __align__(16) float socPT[NROWP * SOC_STRIDE];
    // stage 3: reordered conv31 weights w31A[o][k'=dh*128+i];
    // stage 4/5: reused as y32[o2*39+p] (1248 <= 6176 floats)
    __shared__ __align__(16) float buf[C31 * W31_STRIDE];
    // conv31 output transposed: y16T[p][o]
    __shared__ __align__(16) float y16T[NCELL * Y16_STRIDE];

    const int b   = blockIdx.x;
    const int tid = threadIdx.x;

    const int wave   = tid >> 5;          // 0..2 -> position tile
    const int lane   = tid & 31;
    const int laneLo = lane & 15;
    const int laneHi = lane >> 4;         // 0: K0/K1, 1: K2/K3

    // ---- stage 1: winners -------------------------------------------------
    if (tid < NCELL) winner[tid] = -1;
    __syncthreads();
    if (tid < KNBR) {
        const int vid = nb[b * KNBR + tid];
        const int px  = gp[vid * 2 + 0];
        const int py  = gp[vid * 2 + 1];
        if (px >= 0 && px < GRID_H && py >= 0 && py < GRID_W) {
            atomicMax(&winner[px * GRID_W + py], tid);
        }
    }
    __syncthreads();

    // ---- stage 2a: zero the H-pad rows (rows 0..2 and 42..44) -------------
    for (int e = tid; e < 6 * ENC_DIM; e += 96) {
        const int rr = e >> 7;                       // 0..5
        const int r  = rr < 3 ? rr : rr + 39;        // 0,1,2,42,43,44
        socPT[r * SOC_STRIDE + (e & 127)] = 0.0f;
    }
    // ---- stage 2b: gather winner features (enc[..., -1]) into socPT -------
    for (int e = tid; e < NCELL * ENC_DIM; e += 96) {
        const int c = e >> 7;                        // cell 0..38
        const int i = e & 127;                       // channel
        const int wk = winner[c];
        float v = 0.0f;
        if (wk >= 0) {
            const int vid = nb[b * KNBR + wk];
            v = enc[(vid * ENC_DIM + i) * TSTEPS + (TSTEPS - 1)];
        }
        socPT[(c + 3) * SOC_STRIDE + i] = v;         // row = cell + 3 (pad)
    }
    // ---- stage 2c: pre-stage w31 reordered to k' = dh*128 + i -------------
    for (int o = 0; o < C31; ++o) {
        for (int r = tid; r < 384; r += 96) {
            const int dh = r >> 7;
            const int i  = r & 127;
            buf[o * W31_STRIDE + r] = w31[o * 384 + i * 3 + dh];
        }
    }
    __syncthreads();

    // ---- stage 3: conv3x1 GEMM, K = dh-major, fp32 WMMA --------------------
    {
        const int p  = wave * 16 + laneLo;           // output position (B/C col)
        const int pc = p < NCELL ? p : NCELL - 1;    // clamp padded columns
        const int aBase = laneLo * W31_STRIDE;       // A row o = laneLo

        v8f acc = {};
#pragma unroll
        for (int dh = 0; dh < 3; ++dh) {
            const int aOff = aBase + dh * ENC_DIM;
            const int bOff = (pc + dh * 3) * SOC_STRIDE;
#pragma unroll 4
            for (int kk = 0; kk < ENC_DIM; kk += 4) {
                const int ib = kk + laneHi * 2;
                v2f a  = *(const v2f*)&buf[aOff + ib];     // ds_load_b64
                v2f bf = *(const v2f*)&socPT[bOff + ib];   // ds_load_b64
                acc = __builtin_amdgcn_wmma_f32_16x16x4_f32(
                    false, a, false, bf, (short)0, acc, false, false);
            }
        }

        // C layout: VGPR j -> row o = j + laneHi*8, column = laneLo
#pragma unroll
        for (int j = 0; j < 8; ++j) {
            const int o = j + laneHi * 8;
            if (p < NCELL) {
                y16T[p * Y16_STRIDE + o] = lrelu(acc[j] + b31[o]);
            }
        }
    }
    __syncthreads();   // all w31A reads done; y16T complete

    // ---- stage 4: conv1x1 16->32 via WMMA (K=16, 2 row tiles) -------------
    {
        const int p  = wave * 16 + laneLo;
        const int pc = p < NCELL ? p : NCELL - 1;

        v8f acc0 = {};   // o2 = 0..15
        v8f acc1 = {};   // o2 = 16..31
#pragma unroll
        for (int kk = 0; kk < C31; kk += 4) {
            const int kb = kk + laneHi * 2;
            v2f bf = *(const v2f*)&y16T[pc * Y16_STRIDE + kb];
            v2f a0 = *(const v2f*)&w11[laneLo * C31 + kb];
            v2f a1 = *(const v2f*)&w11[(laneLo + 16) * C31 + kb];
            acc0 = __builtin_amdgcn_wmma_f32_16x16x4_f32(
                false, a0, false, bf, (short)0, acc0, false, false);
            acc1 = __builtin_amdgcn_wmma_f32_16x16x4_f32(
                false, a1, false, bf, (short)0, acc1, false, false);
        }

        float* y32 = buf;   // overlay: w31A is dead now
#pragma unroll
        for (int j = 0; j < 8; ++j) {
            const int o  = j + laneHi * 8;
            if (p < NCELL) {
                y32[o * NCELL + p]        = lrelu(acc0[j] + b11[o]);
                y32[(o + 16) * NCELL + p] = lrelu(acc1[j] + b11[o + 16]);
            }
        }
    }
    __syncthreads();

    // ---- stage 5: 3x3 stride-3 maxpool (H pad (0,2) with -inf) ------------
    // pooled shape (32,5,1), flatten index t = o2*5 + ph
    {
        const float* y32 = buf;
        for (int t = tid; t < FC_DIM; t += 96) {
            const int o2 = t / 5;
            const int ph = t - o2 * 5;
            const int h0 = ph * 3;
            const int h1 = (h0 + 3 < GRID_H) ? h0 + 3 : GRID_H;
            float m = -FLT_MAX;
            for (int h = h0; h < h1; ++h) {
#pragma unroll
                for (int w = 0; w < GRID_W; ++w) {
                    const float v = y32[o2 * NCELL + h * GRID_W + w];
                    m = v > m ? v : m;
                }
            }
            X[b * FC_DIM + t] = m;
        }
    }
}

// ---------------------------------------------------------------------------
// Kernel 2: out = lrelu(X @ fc_w^T + fc_b), (8192x160)x(160x64)
// ---------------------------------------------------------------------------
__global__ __launch_bounds__(128)
void csp_fc_kernel(const float* __restrict__ X,      // (B,160)
                   const float* __restrict__ fcw,    // (64,160)
                   const float* __restrict__ fcb,    // (64)
                   float*       __restrict__ out)    // (B,64)
{
    const int wave   = threadIdx.x >> 5;
    const int lane   = threadIdx.x & 31;
    const int laneLo = lane & 15;
    const int laneHi = lane >> 4;

    const int tileId = blockIdx.x * 4 + wave;      // 2048 tiles
    const int mtile  = tileId >> 2;                // 512 row tiles
    const int ntile  = tileId & 3;                 // 4 col tiles

    const int row = mtile * 16 + laneLo;           // A row this lane owns
    const int col = ntile * 16 + laneLo;           // B column this lane owns

    v8f acc = {};
#pragma unroll 8
    for (int kk = 0; kk < FC_DIM; kk += 4) {
        const int kb = kk + laneHi * 2;
        v2f a  = *(const v2f*)&X[row * FC_DIM + kb];
        v2f bf = *(const v2f*)&fcw[col * FC_DIM + kb];
        acc = __builtin_amdgcn_wmma_f32_16x16x4_f32(
            false, a, false, bf, (short)0, acc, false, false);
    }

    const float bias = fcb[col];
#pragma unroll
    for (int j = 0; j < 8; ++j) {
        const int m = mtile * 16 + j + laneHi * 8;
        out[m * SOC_D + col] = lrelu(acc[j] + bias);
    }
}

// ---------------------------------------------------------------------------
extern "C" void kernel_launch(void* const* d_in, const int* in_sizes, int n_in,
                              void* d_out, int out_size, void* d_ws, size_t ws_size,
                              hipStream_t stream) {
    const float* enc  = (const float*)d_in[0];   // (100000,128,16)
    const int*   nb   = (const int*)  d_in[1];   // (8192,32)
    const int*   gp   = (const int*)  d_in[2];   // (100000,2)
    const float* w31  = (const float*)d_in[3];   // (16,128,3,1)
    const float* b31  = (const float*)d_in[4];   // (16)
    const float* w11  = (const float*)d_in[5];   // (32,16,1,1)
    const float* b11  = (const float*)d_in[6];   // (32)
    const float* fcw  = (const float*)d_in[7];   // (64,160)
    const float* fcb  = (const float*)d_in[8];   // (64)
    float*       out  = (float*)d_out;           // (8192,64)
    float*       Xact = (float*)d_ws;            // (8192,160) activations

    csp_grid_conv_kernel<<<BATCH, 96, 0, stream>>>(enc, nb, gp, w31, b31,
                                                   w11, b11, Xact);
    csp_fc_kernel<<<512, 128, 0, stream>>>(Xact, fcw, fcb, out);
}